// GAT_43602507989730
// MI455X (gfx1250) — compile-verified
//
#include <hip/hip_runtime.h>
#include <math.h>

// N=256, H=128, E=64, K=4, ALPHA=0.2
typedef __attribute__((ext_vector_type(2))) float v2f;
typedef __attribute__((ext_vector_type(8))) float v8f;
typedef __attribute__((ext_vector_type(4))) int v4i;

#define NN 256
#define HH 128
#define EE 64
#define KK 4

// ---- typed address-space pointers for the gfx1250 async-to-LDS builtin ----
// builtin signature (from clang diagnostic): (int4 AS1*, int4 AS3*, Ii, Ii)
typedef __attribute__((address_space(1))) v4i* g_v4i_p;
typedef __attribute__((address_space(3))) v4i* l_v4i_p;

__device__ __forceinline__ g_v4i_p gbl_cast4(const void* p) {
  return (g_v4i_p)(unsigned long long)p;
}
__device__ __forceinline__ l_v4i_p lds_cast4(const void* p) {
  // generic LDS pointer -> 32-bit LDS offset (flat LDS addr = addr[31:0])
  return (l_v4i_p)(unsigned int)(unsigned long long)p;
}

// Cooperatively copy 128x128 f32 (64KB) global -> LDS with async b128 ops.
// 128 threads: each issues 32 async transfers; wave waits its ASYNCcnt,
// then the block barrier makes the tile visible to all waves.
__device__ __forceinline__ void stage_w_async(const float* __restrict__ gsrc,
                                              float* lds_dst, int tid) {
#pragma unroll
  for (int it = 0; it < 32; ++it) {
    int idx4 = it * 128 + tid;  // b128 index
    __builtin_amdgcn_global_load_async_to_lds_b128(
        gbl_cast4(gsrc + idx4 * 4), lds_cast4(lds_dst + idx4 * 4), 0, 0);
  }
  asm volatile("s_wait_asynccnt 0x0" ::: "memory");
  __syncthreads();
}

// ---------------------------------------------------------------------------
// dist kernel: replicates cal_dist() exactly (including its quirks).
// ---------------------------------------------------------------------------
__global__ void dist_kernel(const float* __restrict__ goal,
                            const float* __restrict__ action,
                            float* __restrict__ dist,
                            float* __restrict__ base) {
  int i = blockIdx.x;
  int j = threadIdx.x;
  float axi = action[i * 2 + 0], ayi = action[i * 2 + 1];
  float gxi = goal[i * 2 + 0],   gyi = goal[i * 2 + 1];
  float bi = sqrtf((axi - gxi) * (axi - gxi) + (ayi - gyi) * (ayi - gyi));
  if (j == 0) base[i] = bi;

  float outv;
  if (i == j) {
    outv = bi;
  } else {
    int r = (i < j) ? i : j;
    int c = (i < j) ? j : i;
    float a1x = action[r * 2 + 0], a1y = action[r * 2 + 1];
    float a2x = action[c * 2 + 0], a2y = action[c * 2 + 1];
    float g1x = goal[r * 2 + 0],   g1y = goal[r * 2 + 1];
    float g2x = goal[c * 2 + 0],   g2y = goal[c * 2 + 1];
    // dn = (a1x-a1x)*(g1y-g2y) - (a1y-a2y)*(g1x-g2x)  -> first term is 0
    float dn = -(a1y - a2y) * (g1x - g2x);
    float cross_a = a1x * a2y - a1y * a2x;
    float gterm = g1x * g2y - g1x * g2x;          // reference quirk kept
    float n1 = cross_a * (g1x - g2x) - (a1x - a2x) * gterm;
    float n2 = cross_a * (g1y - g2y) - (a1y - a2x) * gterm;  // quirk kept
    float safe = (dn == 0.0f) ? 1.0f : dn;
    float p1 = n1 / safe, p2 = n2 / safe;
    bool cond = (dn != 0.0f) && ((a1x - p1) * (p1 - g1x) > 0.0f);
    if (i < j) {
      outv = cond ? sqrtf((a1x - p1) * (a1x - p1) + (a1y - p2) * (a1y - p2)) : bi;
    } else {
      outv = cond ? sqrtf((a2x - p1) * (a2x - p1) + (a2y - p2) * (a2y - p2)) : bi;
    }
  }
  dist[i * NN + j] = outv;
}

// ---------------------------------------------------------------------------
// prep kernel (1 block, 128 threads):
//  u = W_emb @ W3, v = W_emb @ W4, cvec = b_emb@(W3+W4) + b_gate
//  wsrc[k] = W_gat[k] @ a_src[k], wdst[k] = W_gat[k] @ a_dst[k]
// ---------------------------------------------------------------------------
__global__ void prep_kernel(const float* __restrict__ W_emb,
                            const float* __restrict__ b_emb,
                            const float* __restrict__ W_gate,
                            const float* __restrict__ b_gate,
                            const float* __restrict__ W_gat,
                            const float* __restrict__ a_src,
                            const float* __restrict__ a_dst,
                            float* __restrict__ u, float* __restrict__ v,
                            float* __restrict__ cvec,
                            float* __restrict__ wsrc, float* __restrict__ wdst) {
  int o = threadIdx.x;  // 0..127
  float su = 0.f, sv = 0.f, sc = 0.f;
  for (int e = 0; e < EE; ++e) {
    float w3 = W_gate[(2 * HH + e) * HH + o];
    float w4 = W_gate[(2 * HH + EE + e) * HH + o];
    su += W_emb[e] * w3;
    sv += W_emb[e] * w4;
    sc += b_emb[e] * (w3 + w4);
  }
  u[o] = su;
  v[o] = sv;
  cvec[o] = sc + b_gate[o];
  for (int k = 0; k < KK; ++k) {
    const float* Wk = W_gat + k * HH * HH + o * HH;  // row f=o of W_gat[k]
    float s1 = 0.f, s2 = 0.f;
    for (int oo = 0; oo < HH; ++oo) {
      float w = Wk[oo];
      s1 += w * a_src[k * HH + oo];
      s2 += w * a_dst[k * HH + oo];
    }
    wsrc[k * HH + o] = s1;
    wdst[k * HH + o] = s2;
  }
}

// ---------------------------------------------------------------------------
// fp32 WMMA GEMM: Y(256x128) = X(256x128) @ W(128x128), row-major, ld==128.
// W is staged into LDS once per block via async-to-LDS; B fragments then come
// from LDS (ds loads) instead of strided VMEM. One wave per 16x16 tile.
// ---------------------------------------------------------------------------
__global__ void wmma_gemm_f32(const float* __restrict__ X,
                              const float* __restrict__ W,
                              float* __restrict__ Y) {
  __shared__ float sW[HH * HH];  // 64 KB
  const int Kd = HH, Ncols = HH;
  int tid = threadIdx.x;
  stage_w_async(W, sW, tid);

  int wave = blockIdx.x * (blockDim.x >> 5) + (tid >> 5);
  int lane = tid & 31;
  const int tiles_n = Ncols / 16;
  int tm = wave / tiles_n;
  int tn = wave % tiles_n;
  int half = lane >> 4;
  int l = lane & 15;
  v8f acc = {};
  const float* xr = X + (tm * 16 + l) * Kd;
  const float* wc = sW + tn * 16 + l;
#pragma unroll 4
  for (int kk = 0; kk < Kd; kk += 4) {
    int ka = kk + 2 * half;
    v2f a;
    a.x = xr[ka];
    a.y = xr[ka + 1];
    v2f b;
    b.x = wc[ka * Ncols];
    b.y = wc[(ka + 1) * Ncols];
    acc = __builtin_amdgcn_wmma_f32_16x16x4_f32(false, a, false, b, (short)0,
                                                acc, false, false);
  }
  int row = tm * 16 + half * 8;
  int col = tn * 16 + l;
#pragma unroll
  for (int r = 0; r < 8; ++r) Y[(row + r) * Ncols + col] = acc[r];
}

// ---------------------------------------------------------------------------
// attention kernel: one block per b (256 blocks x 256 threads).
// Produces CTX[k][b][o] = sum_j alpha[b,k,j] * gat_in[b,j,o]  (j includes the
// goal slot j=256). Logits use leaky_relu(src_scalar + dst), softmax over 257.
// ---------------------------------------------------------------------------
__global__ void attn_kernel(const float* __restrict__ ha,
                            const float* __restrict__ hg,
                            const float* __restrict__ A,
                            const float* __restrict__ Bm,
                            const float* __restrict__ Cm,
                            const float* __restrict__ dist,
                            const float* __restrict__ base,
                            const float* __restrict__ u,
                            const float* __restrict__ v,
                            const float* __restrict__ cvec,
                            const float* __restrict__ wsrc,
                            const float* __restrict__ wdst,
                            float* __restrict__ CTX) {
  int b = blockIdx.x;
  int tid = threadIdx.x;  // 0..255

  __shared__ float sA[HH], sU[HH], sV[HH], sC[HH], sHA[HH], sCG[HH];
  __shared__ float sWd[KK][HH];
  __shared__ float sLog[KK][NN + 1];
  __shared__ float sS[KK];
  __shared__ float red[NN];
  __shared__ float sCtx[2][KK][HH];

  // ---- phase 0: per-feature loads + self-gate (curr_goal) ----
  if (tid < HH) {
    int o = tid;
    float Ab = A[b * HH + o];
    float uo = u[o], vo = v[o], co = cvec[o];
    sA[o] = Ab;
    sU[o] = uo;
    sV[o] = vo;
    sC[o] = co;
    sHA[o] = ha[b * HH + o];
    float pre = Ab + Cm[b * HH + o] + base[b] * (uo + vo) + co;
    sCG[o] = hg[b * HH + o] * (1.0f / (1.0f + expf(-pre)));
    for (int k = 0; k < KK; ++k) sWd[k][o] = wdst[k * HH + o];
  }
  __syncthreads();

  // ---- phase 0b: src scalar s_k = ha[b].wsrc_k ; goal logit = cg.wdst_k ----
  for (int k = 0; k < KK; ++k) {
    if (tid < HH) {
      red[tid] = sHA[tid] * wsrc[k * HH + tid];
      red[HH + tid] = sCG[tid] * sWd[k][tid];
    }
    __syncthreads();
    for (int s = 64; s > 0; s >>= 1) {
      if (tid < s)
        red[tid] += red[tid + s];
      else if (tid >= HH && tid < HH + s)
        red[tid] += red[tid + s];
      __syncthreads();
    }
    if (tid == 0) {
      sS[k] = red[0];
      float e = red[0] + red[HH];
      sLog[k][NN] = (e > 0.0f) ? e : 0.2f * e;  // leaky_relu
    }
    __syncthreads();
  }

  // ---- phase 1: per-j gate + dst dot products -> raw leaky-relu logits ----
  {
    int j = tid;
    float dbj = dist[b * NN + j];
    float djb = dist[j * NN + b];
    float a0 = 0.f, a1 = 0.f, a2 = 0.f, a3 = 0.f;
    for (int o = 0; o < HH; ++o) {
      float gp = sA[o] + Bm[j * HH + o] + dbj * sU[o] + djb * sV[o] + sC[o];
      float g = 1.0f / (1.0f + expf(-gp));
      float ca = (j == b) ? sHA[o] : ha[j * HH + o] * g;
      a0 += ca * sWd[0][o];
      a1 += ca * sWd[1][o];
      a2 += ca * sWd[2][o];
      a3 += ca * sWd[3][o];
    }
    float e;
    e = sS[0] + a0; sLog[0][j] = (e > 0.f) ? e : 0.2f * e;
    e = sS[1] + a1; sLog[1][j] = (e > 0.f) ? e : 0.2f * e;
    e = sS[2] + a2; sLog[2][j] = (e > 0.f) ? e : 0.2f * e;
    e = sS[3] + a3; sLog[3][j] = (e > 0.f) ? e : 0.2f * e;
  }
  __syncthreads();

  // ---- phase 2: softmax over 257 entries, per k (alpha stored in sLog) ----
  for (int k = 0; k < KK; ++k) {
    float lj = sLog[k][tid];
    red[tid] = lj;
    __syncthreads();
    if (tid == 0) red[0] = fmaxf(red[0], sLog[k][NN]);
    __syncthreads();
    for (int s = 128; s > 0; s >>= 1) {
      if (tid < s) red[tid] = fmaxf(red[tid], red[tid + s]);
      __syncthreads();
    }
    float m = red[0];
    __syncthreads();
    float ev = expf(lj - m);
    red[tid] = ev;
    __syncthreads();
    if (tid == 0) {
      float e2 = expf(sLog[k][NN] - m);
      sLog[k][NN] = e2;
      red[0] += e2;
    }
    __syncthreads();
    for (int s = 128; s > 0; s >>= 1) {
      if (tid < s) red[tid] += red[tid + s];
      __syncthreads();
    }
    float ssum = red[0];
    __syncthreads();
    sLog[k][tid] = ev / ssum;
    if (tid == 0) sLog[k][NN] = sLog[k][NN] / ssum;
    __syncthreads();
  }

  // ---- phase 3: context vectors ctx[k][o] = sum_j alpha[k][j]*ca[j][o] ----
  {
    int o = tid & (HH - 1);
    int half = tid >> 7;  // 0 or 1
    float c0 = 0.f, c1 = 0.f, c2 = 0.f, c3 = 0.f;
    for (int jj = 0; jj < 128; ++jj) {
      int j = half * 128 + jj;
      float dbj = dist[b * NN + j];
      float djb = dist[j * NN + b];
      float gp = sA[o] + Bm[j * HH + o] + dbj * sU[o] + djb * sV[o] + sC[o];
      float g = 1.0f / (1.0f + expf(-gp));
      float ca = (j == b) ? sHA[o] : ha[j * HH + o] * g;
      c0 += sLog[0][j] * ca;
      c1 += sLog[1][j] * ca;
      c2 += sLog[2][j] * ca;
      c3 += sLog[3][j] * ca;
    }
    if (half == 0) {  // goal slot j = 256
      c0 += sLog[0][NN] * sCG[o];
      c1 += sLog[1][NN] * sCG[o];
      c2 += sLog[2][NN] * sCG[o];
      c3 += sLog[3][NN] * sCG[o];
    }
    sCtx[half][0][o] = c0;
    sCtx[half][1][o] = c1;
    sCtx[half][2][o] = c2;
    sCtx[half][3][o] = c3;
    __syncthreads();
    if (half == 0) {
      for (int k = 0; k < KK; ++k)
        CTX[(k * NN + b) * HH + o] = sCtx[0][k][o] + sCtx[1][k][o];
    }
  }
}

// ---------------------------------------------------------------------------
// output kernel: out(256x128) = elu( 0.25 * sum_k CTX_k @ W_gat[k] + bias )
// fp32 WMMA; each W_gat[k] slice is async-staged into LDS per head iteration.
// ---------------------------------------------------------------------------
__global__ void wmma_out_f32(const float* __restrict__ CTX,
                             const float* __restrict__ W_gat,
                             const float* __restrict__ gat_bias,
                             float* __restrict__ out) {
  __shared__ float sW[HH * HH];  // 64 KB, reused per head
  const int Kd = HH, Ncols = HH;
  int tid = threadIdx.x;
  int wave = blockIdx.x * (blockDim.x >> 5) + (tid >> 5);
  int lane = tid & 31;
  const int tiles_n = Ncols / 16;
  int tm = wave / tiles_n;
  int tn = wave % tiles_n;
  int half = lane >> 4;
  int l = lane & 15;
  v8f acc = {};
  for (int k = 0; k < KK; ++k) {
    stage_w_async(W_gat + k * HH * HH, sW, tid);
    const float* xr = CTX + k * NN * HH + (tm * 16 + l) * Kd;
    const float* wc = sW + tn * 16 + l;
#pragma unroll 4
    for (int kk = 0; kk < Kd; kk += 4) {
      int ka = kk + 2 * half;
      v2f a;
      a.x = xr[ka];
      a.y = xr[ka + 1];
      v2f b;
      b.x = wc[ka * Ncols];
      b.y = wc[(ka + 1) * Ncols];
      acc = __builtin_amdgcn_wmma_f32_16x16x4_f32(false, a, false, b, (short)0,
                                                  acc, false, false);
    }
    __syncthreads();  // protect sW before next head restages it
  }
  int row = tm * 16 + half * 8;
  int col = tn * 16 + l;
#pragma unroll
  for (int r = 0; r < 8; ++r) {
    float vv = acc[r] * 0.25f + gat_bias[col];
    vv = (vv > 0.0f) ? vv : (expf(vv) - 1.0f);  // elu(alpha=1)
    out[(row + r) * HH + col] = vv;
  }
}

// ---------------------------------------------------------------------------
extern "C" void kernel_launch(void* const* d_in, const int* in_sizes, int n_in,
                              void* d_out, int out_size, void* d_ws,
                              size_t ws_size, hipStream_t stream) {
  const float* ha = (const float*)d_in[0];        // (256,128)
  const float* hg = (const float*)d_in[1];        // (256,128)
  const float* goal = (const float*)d_in[2];      // (256,2)
  const float* action = (const float*)d_in[3];    // (256,2)
  const float* W_emb = (const float*)d_in[4];     // (1,64)
  const float* b_emb = (const float*)d_in[5];     // (64,)
  const float* W_gate = (const float*)d_in[6];    // (384,128)
  const float* b_gate = (const float*)d_in[7];    // (128,)
  const float* W_gat = (const float*)d_in[8];     // (4,128,128)
  const float* a_src = (const float*)d_in[9];     // (4,128)
  const float* a_dst = (const float*)d_in[10];    // (4,128)
  const float* gat_bias = (const float*)d_in[11]; // (128,)
  float* out = (float*)d_out;                     // (256,128)

  float* ws = (float*)d_ws;
  float* dist = ws;                   // 65536
  float* base = dist + NN * NN;       // 256
  float* A = base + NN;               // 32768
  float* Bm = A + NN * HH;            // 32768
  float* Cm = Bm + NN * HH;           // 32768
  float* u = Cm + NN * HH;            // 128
  float* v = u + HH;                  // 128
  float* cvec = v + HH;               // 128
  float* wsrc = cvec + HH;            // 512
  float* wdst = wsrc + KK * HH;       // 512
  float* CTX = wdst + KK * HH;        // 131072

  dist_kernel<<<NN, NN, 0, stream>>>(goal, action, dist, base);
  prep_kernel<<<1, HH, 0, stream>>>(W_emb, b_emb, W_gate, b_gate, W_gat, a_src,
                                    a_dst, u, v, cvec, wsrc, wdst);
  // A = ha @ W1, B = ha @ W2, C = hg @ W2   (W_gate rows [0:128) and [128:256))
  wmma_gemm_f32<<<32, 128, 0, stream>>>(ha, W_gate, A);
  wmma_gemm_f32<<<32, 128, 0, stream>>>(ha, W_gate + HH * HH, Bm);
  wmma_gemm_f32<<<32, 128, 0, stream>>>(hg, W_gate + HH * HH, Cm);
  attn_kernel<<<NN, NN, 0, stream>>>(ha, hg, A, Bm, Cm, dist, base, u, v, cvec,
                                     wsrc, wdst, CTX);
  wmma_out_f32<<<32, 128, 0, stream>>>(CTX, W_gat, gat_bias, out);
}